// Repro_91070486545038
// MI455X (gfx1250) — compile-verified
//
#include <hip/hip_runtime.h>

// out = 2.0f * tangents_1 + 99.0f * full_default   (elementwise, f32)
// 100*2048*512 = 104,857,600 elements -> 1.258 GB moved/call.
// HBM roofline @23.3 TB/s ~= 54 us; FLOP side trivial (2 flop/elem).
// Strategy: guard-free 128-bit non-temporal streaming, wave32 blocks,
// 32-bit GVS addressing (byte offsets < 2^31).

typedef __attribute__((ext_vector_type(4))) float v4f;

#define THREADS 256
#define CHUNKS_PER_THREAD 4   // 4 x float4 = 64 B per input per thread

// Hot kernel: every block is fully covered -> no bounds checks, no exec
// masking, no zero-init of load destinations.
__global__ __launch_bounds__(THREADS) void fused_axpby_v4_full(
    const v4f* __restrict__ fd,   // full_default as float4
    const v4f* __restrict__ t1,   // tangents_1 as float4
    v4f* __restrict__ out)
{
    // 32-bit chunk index: max byte offset 419 MB < 2^31, so the compiler can
    // use SGPR64-base + VGPR32-offset global addressing (no 64-bit VALU adds).
    const unsigned base =
        blockIdx.x * (unsigned)(THREADS * CHUNKS_PER_THREAD) + threadIdx.x;

    v4f a[CHUNKS_PER_THREAD];
    v4f b[CHUNKS_PER_THREAD];

    // Issue all 8 non-temporal b128 loads back-to-back (zero reuse: working
    // set 1.26 GB >> 192 MB L2, NT keeps it from thrashing L2).
#pragma unroll
    for (int k = 0; k < CHUNKS_PER_THREAD; ++k) {
        a[k] = __builtin_nontemporal_load(&fd[base + k * THREADS]);
        b[k] = __builtin_nontemporal_load(&t1[base + k * THREADS]);
    }

#pragma unroll
    for (int k = 0; k < CHUNKS_PER_THREAD; ++k) {
        v4f r;
#pragma unroll
        for (int c = 0; c < 4; ++c) {
            // r = 99*a + 2*b : one v_mul + one v_fma per component
            r[c] = __builtin_fmaf(99.0f, a[k][c], 2.0f * b[k][c]);
        }
        __builtin_nontemporal_store(r, &out[base + k * THREADS]);
    }
}

// Scalar tail for whatever the full blocks don't cover (never launched for
// this shape, but keeps the launcher fully general).
__global__ __launch_bounds__(64) void fused_axpby_tail(
    const float* __restrict__ fd, const float* __restrict__ t1,
    float* __restrict__ out, long long start, long long n)
{
    long long i = start + (long long)blockIdx.x * blockDim.x + threadIdx.x;
    if (i < n) {
        out[i] = __builtin_fmaf(99.0f, fd[i], 2.0f * t1[i]);
    }
}

extern "C" void kernel_launch(void* const* d_in, const int* in_sizes, int n_in,
                              void* d_out, int out_size, void* d_ws, size_t ws_size,
                              hipStream_t stream) {
    (void)in_sizes; (void)n_in; (void)d_ws; (void)ws_size;

    const float* fd  = (const float*)d_in[0];   // full_default
    const float* t1  = (const float*)d_in[1];   // tangents_1
    float*       out = (float*)d_out;

    const long long n  = (long long)out_size;   // 104,857,600
    const long long n4 = n / 4;                 // 26,214,400 float4 chunks

    const long long per_block   = (long long)THREADS * CHUNKS_PER_THREAD; // 1024
    const long long full_blocks = n4 / per_block;                         // 25,600
    const long long covered     = full_blocks * per_block * 4;            // in floats

    if (full_blocks > 0) {
        fused_axpby_v4_full<<<(int)full_blocks, THREADS, 0, stream>>>(
            (const v4f*)fd, (const v4f*)t1, (v4f*)out);
    }

    const long long rem = n - covered;          // 0 for this shape
    if (rem > 0) {
        const int tb = (int)((rem + 63) / 64);
        fused_axpby_tail<<<tb, 64, 0, stream>>>(fd, t1, out, covered, n);
    }
}